// MultimodalRouting_17282948399418
// MI455X (gfx1250) — compile-verified
//
#include <hip/hip_runtime.h>

typedef __attribute__((ext_vector_type(2))) float v2f;
typedef __attribute__((ext_vector_type(8))) float v8f;

#define D_DIM   1024
#define N_DIM   8
#define C_DIM   4
#define NROWS   (N_DIM * C_DIM)   // 32 (n,c) rows
#define VS      1028              // padded LDS row stride (floats) -> conflict-free WMMA fragment loads
#define NITER   3
#define EPS_F   1e-6f

// ---------------------------------------------------------------------------
// Stage kernel: gather the diagonal W[n,d,c,d] once into wdiag[(n*C+c)*D + d]
// so the fused kernel reads it coalesced (and L2-resident for all 256 blocks).
// ---------------------------------------------------------------------------
__global__ void wdiag_gather_kernel(const float* __restrict__ W,
                                    float* __restrict__ wdiag) {
    int t = blockIdx.x * blockDim.x + threadIdx.x;
    if (t >= NROWS * D_DIM) return;
    int row = t >> 10;            // n*4 + c
    int d   = t & (D_DIM - 1);
    int n = row >> 2, c = row & 3;
    size_t idx = (((size_t)n * D_DIM + d) * C_DIM + c) * D_DIM + d;
    wdiag[t] = W[idx];
}

// ---------------------------------------------------------------------------
// Fused routing kernel: one block per batch b, 256 threads (8 waves).
// votes[32][D] and cj[4][D] live in LDS for all 3 iterations.
// sij GEMM (32 x 1024 x 4) runs on v_wmma_f32_16x16x4_f32:
//   wave w owns K-chunk [128w,128w+128); acc over 32 WMMA steps x 2 row tiles.
// ---------------------------------------------------------------------------
__global__ __launch_bounds__(256)
void routing_kernel(const float* __restrict__ fi, const float* __restrict__ pi,
                    const float* __restrict__ W, const float* __restrict__ wdiag,
                    const float* __restrict__ cj_init,
                    float* __restrict__ out_cj, float* __restrict__ out_rij) {
    extern __shared__ float smem[];
    float* votes = smem;                       // NROWS * VS
    float* cjl   = votes + NROWS * VS;         // C_DIM * VS
    float* sij   = cjl + C_DIM * VS;           // NROWS
    float* nrm   = sij + NROWS;                // C_DIM

    const int b    = blockIdx.x;
    const int tid  = threadIdx.x;
    const int lane = tid & 31;                 // wave32
    const int wave = tid >> 5;

    float pin[N_DIM];
#pragma unroll
    for (int n = 0; n < N_DIM; ++n) pin[n] = pi[b * N_DIM + n];

    if (tid < C_DIM) nrm[tid] = 0.0f;
    if (tid >= 64 && tid < 64 + NROWS) sij[tid - 64] = 0.0f;

    // ---- votes = fi * diag(W), into LDS (thread owns d = tid + 256k) ----
#pragma unroll
    for (int k = 0; k < 4; ++k) {
        int d = tid + 256 * k;
#pragma unroll
        for (int n = 0; n < N_DIM; ++n) {
            float f = fi[((size_t)b * N_DIM + n) * D_DIM + d];
#pragma unroll
            for (int c = 0; c < C_DIM; ++c) {
                int row = n * C_DIM + c;
                float wv = wdiag ? wdiag[row * D_DIM + d]
                                 : W[(((size_t)n * D_DIM + d) * C_DIM + c) * D_DIM + d];
                votes[row * VS + d] = f * wv;
            }
        }
    }

    // ---- cj = l2norm(cj_init) ----
    float cjn[C_DIM][4];
    float psum[C_DIM];
#pragma unroll
    for (int c = 0; c < C_DIM; ++c) {
        psum[c] = 0.0f;
#pragma unroll
        for (int k = 0; k < 4; ++k) {
            int d = tid + 256 * k;
            float v = cj_init[((size_t)b * C_DIM + c) * D_DIM + d];
            cjn[c][k] = v;
            psum[c] += v * v;
        }
    }
    __syncthreads();                            // zeros + votes visible
#pragma unroll
    for (int c = 0; c < C_DIM; ++c) {
        float s = psum[c];
#pragma unroll
        for (int off = 16; off >= 1; off >>= 1) s += __shfl_xor(s, off, 32);
        if (lane == 0) atomicAdd(&nrm[c], s);
    }
    __syncthreads();
#pragma unroll
    for (int c = 0; c < C_DIM; ++c) {
        float inv = 1.0f / (sqrtf(nrm[c]) + EPS_F);
#pragma unroll
        for (int k = 0; k < 4; ++k) {
            int d = tid + 256 * k;
            cjl[c * VS + d] = cjn[c][k] * inv;
        }
    }
    __syncthreads();
    if (tid < C_DIM) nrm[tid] = 0.0f;
    __syncthreads();

    // WMMA fragment addressing (ISA 7.12.2, f32 16x4 A / 4x16 B):
    //   lanes 0-15 carry K={0,1}, lanes 16-31 carry K={2,3} of each 4-wide step.
    const int    arow = lane & 15;
    const int    koff = (lane & 16) ? 2 : 0;
    const float* a0p  = votes + arow * VS;          // tile0: rows 0..15
    const float* a1p  = votes + (16 + arow) * VS;   // tile1: rows 16..31
    const bool   bval = arow < C_DIM;               // B columns j>=4 are zero
    const float* bp   = cjl + (bval ? arow : 0) * VS;
    const int    kbase = wave * 128;                // per-wave K chunk

    float wgt[N_DIM][C_DIM];

    for (int it = 0; it < NITER; ++it) {
        // ---- sij[m, j] = sum_d votes[m][d] * cj[j][d] via WMMA ----
        v8f acc0 = {0.f, 0.f, 0.f, 0.f, 0.f, 0.f, 0.f, 0.f};
        v8f acc1 = acc0;
#pragma unroll 4
        for (int s = 0; s < 32; ++s) {
            int d0 = kbase + 4 * s + koff;
            v2f a0 = *(const v2f*)(a0p + d0);
            v2f a1 = *(const v2f*)(a1p + d0);
            v2f bb = *(const v2f*)(bp + d0);
            v2f bz = {0.f, 0.f};
            bb = bval ? bb : bz;
            acc0 = __builtin_amdgcn_wmma_f32_16x16x4_f32(false, a0, false, bb,
                                                         (short)0, acc0, false, false);
            acc1 = __builtin_amdgcn_wmma_f32_16x16x4_f32(false, a1, false, bb,
                                                         (short)0, acc1, false, false);
        }
        // D layout: VGPR v, lanes0-15 -> (M=v, N=lane); lanes16-31 -> (M=v+8, N=lane-16).
        // We need N == M%4; combine the 8 wave-partials with LDS atomics.
#pragma unroll
        for (int v = 0; v < 8; ++v) {
            if ((lane & 15) == (v & 3)) {
                int mrow = v + ((lane >> 4) << 3);
                atomicAdd(&sij[mrow],      acc0[v]);
                atomicAdd(&sij[16 + mrow], acc1[v]);
            }
        }
        __syncthreads();                        // sij complete

        // ---- softmax over C=4, weighted = pi * rij (block-uniform) ----
#pragma unroll
        for (int n = 0; n < N_DIM; ++n) {
            float s0 = sij[n * 4 + 0], s1 = sij[n * 4 + 1];
            float s2 = sij[n * 4 + 2], s3 = sij[n * 4 + 3];
            float mx = fmaxf(fmaxf(s0, s1), fmaxf(s2, s3));
            float e0 = __expf(s0 - mx), e1 = __expf(s1 - mx);
            float e2 = __expf(s2 - mx), e3 = __expf(s3 - mx);
            float rs = pin[n] / (e0 + e1 + e2 + e3);
            wgt[n][0] = e0 * rs; wgt[n][1] = e1 * rs;
            wgt[n][2] = e2 * rs; wgt[n][3] = e3 * rs;
        }

        // ---- cj_new[c][d] = sum_n wgt[n][c] * votes[n*4+c][d] ----
#pragma unroll
        for (int c = 0; c < C_DIM; ++c) {
            psum[c] = 0.0f;
#pragma unroll
            for (int k = 0; k < 4; ++k) {
                int d = tid + 256 * k;
                float a = 0.0f;
#pragma unroll
                for (int n = 0; n < N_DIM; ++n)
                    a = fmaf(wgt[n][c], votes[(n * C_DIM + c) * VS + d], a);
                cjn[c][k] = a;
                psum[c] += a * a;
            }
        }
#pragma unroll
        for (int c = 0; c < C_DIM; ++c) {
            float s = psum[c];
#pragma unroll
            for (int off = 16; off >= 1; off >>= 1) s += __shfl_xor(s, off, 32);
            if (lane == 0) atomicAdd(&nrm[c], s);
        }
        __syncthreads();                        // norms done; sij reads done

        // ---- l2 normalize; publish cj for next iteration ----
#pragma unroll
        for (int c = 0; c < C_DIM; ++c) {
            float inv = 1.0f / (sqrtf(nrm[c]) + EPS_F);
#pragma unroll
            for (int k = 0; k < 4; ++k) {
                int d = tid + 256 * k;
                float v = cjn[c][k] * inv;
                cjn[c][k] = v;
                if (it < NITER - 1) cjl[c * VS + d] = v;
            }
        }
        __syncthreads();                        // nrm reads + cjl writes done
        if (tid < C_DIM) nrm[tid] = 0.0f;
        if (it < NITER - 1 && tid >= 64 && tid < 64 + NROWS) sij[tid - 64] = 0.0f;
        __syncthreads();
    }

    // ---- outputs: cj (B,C,D) then rij_last (B,N,C) ----
#pragma unroll
    for (int c = 0; c < C_DIM; ++c)
#pragma unroll
        for (int k = 0; k < 4; ++k) {
            int d = tid + 256 * k;
            out_cj[((size_t)b * C_DIM + c) * D_DIM + d] = cjn[c][k];
        }
    if (tid < NROWS) {                          // sij of last iter was preserved
        int n = tid >> 2;
        float s0 = sij[n * 4 + 0], s1 = sij[n * 4 + 1];
        float s2 = sij[n * 4 + 2], s3 = sij[n * 4 + 3];
        float mx = fmaxf(fmaxf(s0, s1), fmaxf(s2, s3));
        float e0 = __expf(s0 - mx), e1 = __expf(s1 - mx);
        float e2 = __expf(s2 - mx), e3 = __expf(s3 - mx);
        out_rij[(size_t)b * NROWS + tid] = __expf(sij[tid] - mx) / (e0 + e1 + e2 + e3);
    }
}

extern "C" void kernel_launch(void* const* d_in, const int* in_sizes, int n_in,
                              void* d_out, int out_size, void* d_ws, size_t ws_size,
                              hipStream_t stream) {
    (void)n_in; (void)out_size;
    const float* fi  = (const float*)d_in[0];
    const float* pi  = (const float*)d_in[1];
    const float* W   = (const float*)d_in[2];
    const float* cji = (const float*)d_in[3];
    const int B = in_sizes[0] / (N_DIM * D_DIM);

    float* out_cj  = (float*)d_out;
    float* out_rij = out_cj + (size_t)B * C_DIM * D_DIM;

    const float* wdiag = nullptr;
    const size_t wbytes = (size_t)NROWS * D_DIM * sizeof(float);
    if (ws_size >= wbytes) {
        int total = NROWS * D_DIM;
        wdiag_gather_kernel<<<(total + 255) / 256, 256, 0, stream>>>(W, (float*)d_ws);
        wdiag = (const float*)d_ws;
    }

    const size_t smem = (size_t)(NROWS * VS + C_DIM * VS + NROWS + C_DIM) * sizeof(float);
    (void)hipFuncSetAttribute((const void*)routing_kernel,
                              hipFuncAttributeMaxDynamicSharedMemorySize, (int)smem);
    routing_kernel<<<B, 256, smem, stream>>>(fi, pi, W, wdiag, cji, out_cj, out_rij);
}